// WBlock_31078383353999
// MI455X (gfx1250) — compile-verified
//
#include <hip/hip_runtime.h>
#include <math.h>

typedef __attribute__((ext_vector_type(2))) float v2f;
typedef __attribute__((ext_vector_type(8))) float v8f;

#define B_    8
#define C_    128
#define HW    256
#define NP    4096        // 64*64 pooled tokens (viewed rows)
#define NTOK  65536       // 256*256 full-res viewed rows
#define BATCH_FP   (NP * C_)      // 524288 floats per batch (fp view)
#define BATCH_IMG  (NTOK * C_)    // 8388608 floats per batch (images view)
#define NSLICE 16                 // token slices for the Gram reduction
#define CC     (C_ * C_)          // 16384

// Pair-interleaved LDS index: rows (k, k+1) of column c are adjacent dwords.
// Fragment (k even, c) = contiguous 8B -> single ds_load_b64 into a VGPR pair.
#define PIDX(k, c) ((((k) >> 1) * C_ + (c)) * 2 + ((k) & 1))

static __device__ __forceinline__ v8f wmma_f32(v2f a, v2f b, v8f c) {
    return __builtin_amdgcn_wmma_f32_16x16x4_f32(false, a, false, b,
                                                 (short)0, c, false, false);
}

// ---------------------------------------------------------------------------
// Kernel 1: 4x4 max pool. Output flat layout [B,C,64,64] == fp view [B,4096,128].
// ---------------------------------------------------------------------------
__global__ void maxpool_k(const float* __restrict__ img, float* __restrict__ fp) {
    int idx = blockIdx.x * 256 + threadIdx.x;       // 0 .. 4,194,303
    int w  = idx & 63;
    int h  = (idx >> 6) & 63;
    int bc = idx >> 12;                              // fused b*C+c, 0..1023
    const float* base = img + ((size_t)bc * HW + h * 4) * HW + w * 4;
    float m = -INFINITY;
#pragma unroll
    for (int r = 0; r < 4; ++r) {
        float4 v = *(const float4*)(base + r * HW);
        m = fmaxf(m, fmaxf(fmaxf(v.x, v.y), fmaxf(v.z, v.w)));
    }
    fp[idx] = m;
}

// ---------------------------------------------------------------------------
// Kernel 2a: partial Gram. grid (NSLICE, B_); each block reduces 256 tokens
// into Gpart[b][slice] (128x128). 8 waves, wave wv owns a 16-row strip.
// ---------------------------------------------------------------------------
__global__ void gram_part_k(const float* __restrict__ fp, float* __restrict__ Gpart) {
    __shared__ float sFP[64 * C_];                   // 32 KB chunk: 64 token rows
    const int sl   = blockIdx.x;
    const int b    = blockIdx.y;
    const float* A = fp + (size_t)b * BATCH_FP + (size_t)sl * (NP / NSLICE) * C_;
    const int tid  = threadIdx.x;
    const int wv   = tid >> 5;
    const int lane = tid & 31;
    const int lh   = lane >> 4;                      // half-wave select (K pair)
    const int lm   = lane & 15;
    const int c0   = wv * 16;

    v8f acc[8] = {};

    for (int chunk = 0; chunk < NP / NSLICE; chunk += 64) {
        __syncthreads();
        // cooperative stage: 64 rows x 128 floats, scattered pair-interleaved
        const float4* src = (const float4*)(A + (size_t)chunk * C_);
        for (int i = tid; i < 64 * C_ / 4; i += 256) {
            float4 v = src[i];
            int r  = i >> 5;                          // 32 float4 per row
            int c4 = (i & 31) * 4;
            int p  = PIDX(r, c4);
            sFP[p]     = v.x;
            sFP[p + 2] = v.y;
            sFP[p + 4] = v.z;
            sFP[p + 6] = v.w;
        }
        __syncthreads();

        for (int j = 0; j < 64; j += 4) {
            const int kr = j + 2 * lh;               // even: this half's K pair
            v2f bf[8];
#pragma unroll
            for (int t = 0; t < 8; ++t)
                bf[t] = *(const v2f*)&sFP[PIDX(kr, t * 16 + lm)];
            const v2f afrag = *(const v2f*)&sFP[PIDX(kr, c0 + lm)];
#pragma unroll
            for (int t = 0; t < 8; ++t)
                acc[t] = wmma_f32(afrag, bf[t], acc[t]);
        }
    }

    float* Gb = Gpart + ((size_t)b * NSLICE + sl) * CC;
#pragma unroll
    for (int t = 0; t < 8; ++t)
#pragma unroll
        for (int r = 0; r < 8; ++r) {
            int row = c0 + r + 8 * lh;
            int col = t * 16 + lm;
            Gb[row * C_ + col] = acc[t][r];
        }
}

// Kernel 2b: deterministic reduction of NSLICE partials. 131072 threads.
__global__ void gram_reduce_k(const float* __restrict__ Gpart, float* __restrict__ G) {
    const int g   = blockIdx.x * 256 + threadIdx.x;  // 0 .. B_*CC-1
    const int b   = g >> 14;
    const int off = g & (CC - 1);
    const float* p = Gpart + (size_t)b * NSLICE * CC + off;
    float s = 0.f;
#pragma unroll
    for (int i = 0; i < NSLICE; ++i) s += p[i * CC];
    G[g] = s;
}

// ---------------------------------------------------------------------------
// Tiny 128x128 stages, tiled: grid (4 d-chunks, B_), 128 threads.
// ---------------------------------------------------------------------------
// T[c][d] = sum_e G[c][e] * Wk[d][e]
__global__ void attn1_k(const float* __restrict__ G, const float* __restrict__ Wk,
                        float* __restrict__ T) {
    const int b = blockIdx.y, dc = blockIdx.x * 32, c = threadIdx.x;
    const float* grow = G + b * CC + c * C_;
    float* Tb = T + b * CC;
    for (int d = 0; d < 32; ++d) {
        const float* wrow = Wk + (dc + d) * C_;
        float a = 0.f;
        for (int e = 0; e < C_; ++e) a += grow[e] * wrow[e];
        Tb[c * C_ + dc + d] = a;
    }
}

// S[c][d] = (1/sqrt(128)) * sum_e Wq[c][e] * T[e][d]
__global__ void attn2_k(const float* __restrict__ Wq, const float* __restrict__ T,
                        float* __restrict__ S) {
    const int b = blockIdx.y, dc = blockIdx.x * 32, c = threadIdx.x;
    const float* Tb = T + b * CC;
    const float* wq = Wq + c * C_;
    float s[32];
#pragma unroll
    for (int d = 0; d < 32; ++d) s[d] = 0.f;
    for (int e = 0; e < C_; ++e) {
        const float w = wq[e];
        const float* trow = Tb + e * C_ + dc;        // uniform across threads
#pragma unroll
        for (int d = 0; d < 32; ++d) s[d] += w * trow[d];
    }
    const float scale = 0.08838834764831845f;        // 1/sqrt(128)
    float* srow = S + b * CC + c * C_ + dc;
#pragma unroll
    for (int d = 0; d < 32; ++d) srow[d] = s[d] * scale;
}

// attn[c][:] = softmax(S[c][:]) ; one row per thread (cheap: 3*128 flops)
__global__ void softmax_k(const float* __restrict__ S, float* __restrict__ attn) {
    const int b = blockIdx.x, c = threadIdx.x;
    const float* srow = S + b * CC + c * C_;
    float v[C_];
    float mx = -INFINITY;
#pragma unroll
    for (int d = 0; d < C_; ++d) { v[d] = srow[d]; mx = fmaxf(mx, v[d]); }
    float sum = 0.f;
#pragma unroll
    for (int d = 0; d < C_; ++d) { v[d] = expf(v[d] - mx); sum += v[d]; }
    const float inv = 1.f / sum;
    float* arow = attn + b * CC + c * C_;
#pragma unroll
    for (int d = 0; d < C_; ++d) arow[d] = v[d] * inv;
}

// M[e][d] = sum_c Wv[c][e] * attn[c][d]   (M = Wv^T @ attn)
__global__ void makeM_k(const float* __restrict__ Wv, const float* __restrict__ attn,
                        float* __restrict__ M) {
    const int b = blockIdx.y, dc = blockIdx.x * 32, e = threadIdx.x;
    const float* Ab = attn + b * CC;
    float m[32];
#pragma unroll
    for (int d = 0; d < 32; ++d) m[d] = 0.f;
    for (int c = 0; c < C_; ++c) {
        const float w = Wv[c * C_ + e];
        const float* arow = Ab + c * C_ + dc;        // uniform across threads
#pragma unroll
        for (int d = 0; d < 32; ++d) m[d] += w * arow[d];
    }
    float* mrow = M + b * CC + e * C_ + dc;
#pragma unroll
    for (int d = 0; d < 32; ++d) mrow[d] = m[d];
}

// ---------------------------------------------------------------------------
// Kernel 4: out[b] = images_view[b] (65536x128) @ M[b] (128x128), f32 WMMA.
// grid (256, 8); block 256 (8 waves); wave computes a 32x128 strip with 16
// accumulators. M staged in LDS pair-interleaved -> 1 ds_load_b64 per B-frag.
// B-frags batched before the WMMA burst for load/compute overlap.
// ---------------------------------------------------------------------------
__global__ void __launch_bounds__(256)
outgemm_k(const float* __restrict__ img, const float* __restrict__ M,
          float* __restrict__ out) {
    __shared__ float sM[CC];                         // 64 KB
    const int b = blockIdx.y;
    const float* Ab = img + (size_t)b * BATCH_IMG;
    float*       Ob = out + (size_t)b * BATCH_IMG;
    const float* Mb = M + b * CC;
    const int tid = threadIdx.x;

    for (int i = tid; i < CC / 4; i += 256) {
        float4 v = ((const float4*)Mb)[i];
        int r  = i >> 5;
        int c4 = (i & 31) * 4;
        int p  = PIDX(r, c4);
        sM[p]     = v.x;
        sM[p + 2] = v.y;
        sM[p + 4] = v.z;
        sM[p + 6] = v.w;
    }
    __syncthreads();

    const int wv = tid >> 5;
    const int lane = tid & 31;
    const int lh = lane >> 4;
    const int lm = lane & 15;
    const int rowbase = blockIdx.x * 256 + wv * 32;

    v8f acc[16] = {};
    const float* a0p = Ab + (size_t)(rowbase + lm) * C_;
    const float* a1p = Ab + (size_t)(rowbase + 16 + lm) * C_;

    for (int k0 = 0; k0 < C_; k0 += 4) {
        const int kr = k0 + 2 * lh;                  // even -> 8B aligned
        v2f a0 = *(const v2f*)(a0p + kr);
        v2f a1 = *(const v2f*)(a1p + kr);
        v2f bf[8];
#pragma unroll
        for (int t = 0; t < 8; ++t)
            bf[t] = *(const v2f*)&sM[PIDX(kr, t * 16 + lm)];
#pragma unroll
        for (int t = 0; t < 8; ++t) {
            acc[t]     = wmma_f32(a0, bf[t], acc[t]);
            acc[t + 8] = wmma_f32(a1, bf[t], acc[t + 8]);
        }
    }

#pragma unroll
    for (int t = 0; t < 8; ++t)
#pragma unroll
        for (int r = 0; r < 8; ++r) {
            const int col  = t * 16 + lm;
            const int row0 = rowbase + r + 8 * lh;
            const int row1 = row0 + 16;
            __builtin_nontemporal_store(acc[t][r],     &Ob[(size_t)row0 * C_ + col]);
            __builtin_nontemporal_store(acc[t + 8][r], &Ob[(size_t)row1 * C_ + col]);
        }
}

// ---------------------------------------------------------------------------
extern "C" void kernel_launch(void* const* d_in, const int* in_sizes, int n_in,
                              void* d_out, int out_size, void* d_ws, size_t ws_size,
                              hipStream_t stream) {
    const float* img = (const float*)d_in[0];
    const float* Wq  = (const float*)d_in[1];
    const float* Wk  = (const float*)d_in[2];
    const float* Wv  = (const float*)d_in[3];
    float* out = (float*)d_out;

    float* ws    = (float*)d_ws;
    float* fp    = ws;                                // 4,194,304 f
    float* Gpart = fp    + (size_t)B_ * BATCH_FP;     // 8*16*16384 = 2,097,152 f
    float* G     = Gpart + (size_t)B_ * NSLICE * CC;  // 131072 f
    float* T     = G     + B_ * CC;
    float* S     = T     + B_ * CC;
    float* attn  = S     + B_ * CC;
    float* M     = attn  + B_ * CC;                   // total ~27.7 MB

    maxpool_k<<<(B_ * C_ * 64 * 64) / 256, 256, 0, stream>>>(img, fp);
    dim3 ggrid(NSLICE, B_);
    gram_part_k<<<ggrid, 256, 0, stream>>>(fp, Gpart);
    gram_reduce_k<<<(B_ * CC) / 256, 256, 0, stream>>>(Gpart, G);
    dim3 tgrid(4, B_);
    attn1_k <<<tgrid, 128, 0, stream>>>(G, Wk, T);
    attn2_k <<<tgrid, 128, 0, stream>>>(Wq, T, S);
    softmax_k<<<B_, 128, 0, stream>>>(S, attn);
    makeM_k <<<tgrid, 128, 0, stream>>>(Wv, attn, M);
    dim3 grid(NTOK / 256, B_);
    outgemm_k<<<grid, 256, 0, stream>>>(img, M, out);
}